// I2TLoss_27341761806531
// MI455X (gfx1250) — compile-verified
//
#include <hip/hip_runtime.h>
#include <math.h>

typedef __attribute__((ext_vector_type(2))) float v2f;
typedef __attribute__((ext_vector_type(8))) float v8f;

#define C_CLASSES 512
#define D_DIM     256

// ---------------------------------------------------------------------------
// Kernel 1: per-row argmax over logits [N, 512]. One wave32 per row.
// Lane reads logits[row*512 + k*32 + lane] (fully coalesced, 128B/wave/iter),
// then butterfly shuffle reduction with first-index tie-break (jnp.argmax).
// ---------------------------------------------------------------------------
__global__ __launch_bounds__(256) void argmax_rows(const float* __restrict__ logits,
                                                   int* __restrict__ labels, int N) {
    const int wave = threadIdx.x >> 5;
    const int lane = threadIdx.x & 31;
    const int row  = blockIdx.x * 8 + wave;
    if (row >= N) return;

    const float* rp = logits + (size_t)row * C_CLASSES;
    float bestV = -INFINITY;
    int   bestI = 0;
#pragma unroll
    for (int k = 0; k < C_CLASSES / 32; ++k) {
        const int idx = k * 32 + lane;
        const float v = rp[idx];
        if (v > bestV) { bestV = v; bestI = idx; }   // strict >: keeps first occurrence
    }
#pragma unroll
    for (int off = 16; off > 0; off >>= 1) {
        const float ov = __shfl_xor(bestV, off, 32);
        const int   oi = __shfl_xor(bestI, off, 32);
        if (ov > bestV || (ov == bestV && oi < bestI)) { bestV = ov; bestI = oi; }
    }
    if (lane == 0) labels[row] = bestI;
}

// ---------------------------------------------------------------------------
// Kernel 2: zero sums[512*256] + counts[512] each call (graph-replay safe).
// ---------------------------------------------------------------------------
__global__ void zero_ws(float* __restrict__ p, int n) {
    const int i = blockIdx.x * blockDim.x + threadIdx.x;
    if (i < n) p[i] = 0.0f;
}

// ---------------------------------------------------------------------------
// Kernel 3: segment-sum scatter. One wave32 per row of img_feats [N, 256].
// sums is 512KB -> fully L2-resident; atomics never leave L2.
// ---------------------------------------------------------------------------
__global__ __launch_bounds__(256) void scatter_add(const float* __restrict__ img,
                                                   const int* __restrict__ labels,
                                                   float* __restrict__ sums,
                                                   float* __restrict__ counts, int N) {
    const int wave = threadIdx.x >> 5;
    const int lane = threadIdx.x & 31;
    const int row  = blockIdx.x * 8 + wave;
    if (row >= N) return;

    const int label = labels[row];
    const float* rp = img + (size_t)row * D_DIM;
    float* sp = sums + (size_t)label * D_DIM;
#pragma unroll
    for (int k = 0; k < D_DIM / 32; ++k) {
        const int d = k * 32 + lane;
        atomicAdd(&sp[d], rp[d]);   // global_atomic_add_f32, coalesced 128B lines
    }
    if (lane == 0) atomicAdd(&counts[label], 1.0f);
}

// ---------------------------------------------------------------------------
// Kernel 4: finalize. dots[c] = dot(sums[c]/max(cnt,1), text[c]) computed as the
// diagonal of 16x16 WMMA tiles: for each 16-class tile,
//   Dtile = means[c0:c0+16, :] (16x256)  @  text[c0:c0+16, :]^T (256x16)
// via 64 chained V_WMMA_F32_16X16X4_F32 (full fp32 path, no precision loss).
//
// Per the CDNA5 ISA layouts (CDNA5 ISA):
//   A (16x4 f32): lanes 0-15 -> M=lane, {v0,v1} = K{0,1}; lanes 16-31 -> K{2,3}
//   B (4x16 f32): mirrored — lane holds column N=lane&15, {v0,v1} = K{2*hi, 2*hi+1}
//   D (16x16 f32, 8 VGPRs): VGPR g: lanes 0-15 -> M=g,N=lane; lanes 16-31 -> M=8+g
// Diagonal (M==N): lanes 0-7 hold D[l][l] in acc[l]; lanes 24-31 hold
// D[8+l-24][...] in acc[lane-24].
// ---------------------------------------------------------------------------
__global__ __launch_bounds__(512) void finalize(const float* __restrict__ sums,
                                                const float* __restrict__ counts,
                                                const float* __restrict__ text,
                                                float* __restrict__ out) {
    __shared__ float counts_s[C_CLASSES];
    __shared__ float s_dots;
    __shared__ float s_cnt;

    const int tid = threadIdx.x;
    if (tid < C_CLASSES) counts_s[tid] = counts[tid];
    if (tid == 0) { s_dots = 0.0f; s_cnt = 0.0f; }
    __syncthreads();

    const int wave = tid >> 5;
    const int lane = tid & 31;
    const int mrow = lane & 15;   // M (row) / N (col) index within tile
    const int hi   = lane >> 4;   // selects K pair {0,1} vs {2,3}

    for (int t = wave; t < C_CLASSES / 16; t += 16) {
        const int c0 = t * 16;
        const float cnt = counts_s[c0 + mrow];
        const float inv = 1.0f / fmaxf(cnt, 1.0f);
        const float* srow = sums + (size_t)(c0 + mrow) * D_DIM;
        const float* trow = text + (size_t)(c0 + mrow) * D_DIM;

        v8f acc = {};
#pragma unroll 8
        for (int k0 = 0; k0 < D_DIM; k0 += 4) {
            const int kk = k0 + 2 * hi;
            v2f a; a.x = srow[kk] * inv; a.y = srow[kk + 1] * inv;  // A = means tile
            v2f b; b.x = trow[kk];       b.y = trow[kk + 1];        // B = text^T tile
            acc = __builtin_amdgcn_wmma_f32_16x16x4_f32(
                /*neg_a=*/false, a, /*neg_b=*/false, b,
                /*c_mod=*/(short)0, acc, /*reuse_a=*/false, /*reuse_b=*/false);
        }

        // Extract diagonal lanes of the 16x16 f32 D tile.
        const bool lo = (lane < 8);
        const bool hi8 = (lane >= 24);
        if (lo || hi8) {
            const int e = lo ? lane : (lane - 24);          // VGPR index within acc
            const int c = c0 + (lo ? lane : (8 + lane - 24)); // class of this diag elem
            float dv;
            switch (e) {
                case 0: dv = acc[0]; break;
                case 1: dv = acc[1]; break;
                case 2: dv = acc[2]; break;
                case 3: dv = acc[3]; break;
                case 4: dv = acc[4]; break;
                case 5: dv = acc[5]; break;
                case 6: dv = acc[6]; break;
                default: dv = acc[7]; break;
            }
            if (counts_s[c] > 0.0f) {
                atomicAdd(&s_dots, dv);   // ds_add_f32
                atomicAdd(&s_cnt, 1.0f);
            }
        }
    }
    __syncthreads();
    if (tid == 0) out[0] = s_dots / fmaxf(s_cnt, 1.0f);
}

// ---------------------------------------------------------------------------
extern "C" void kernel_launch(void* const* d_in, const int* in_sizes, int n_in,
                              void* d_out, int out_size, void* d_ws, size_t ws_size,
                              hipStream_t stream) {
    const float* logits = (const float*)d_in[0];   // [N, 512]
    const float* img    = (const float*)d_in[1];   // [N, 256]
    const float* text   = (const float*)d_in[2];   // [512, 256]
    float* out = (float*)d_out;

    const int N = in_sizes[1] / D_DIM;

    // Workspace layout: labels[N] int | sums[512*256] f32 | counts[512] f32
    char* ws = (char*)d_ws;
    int*   labels = (int*)ws;
    size_t off = ((size_t)N * sizeof(int) + 255) & ~(size_t)255;
    float* sums   = (float*)(ws + off);
    float* counts = sums + (size_t)C_CLASSES * D_DIM;

    const int rowBlocks = (N + 7) / 8;

    argmax_rows<<<rowBlocks, 256, 0, stream>>>(logits, labels, N);

    const int nz = C_CLASSES * D_DIM + C_CLASSES;
    zero_ws<<<(nz + 255) / 256, 256, 0, stream>>>(sums, nz);

    scatter_add<<<rowBlocks, 256, 0, stream>>>(img, labels, sums, counts, N);

    finalize<<<1, 512, 0, stream>>>(sums, counts, text, out);
}